// ViPC_73538430042252
// MI455X (gfx1250) — compile-verified
//
#include <hip/hip_runtime.h>

typedef __attribute__((ext_vector_type(2))) float v2f;
typedef __attribute__((ext_vector_type(8))) float v8f;

#define BATCH   32
#define NCLOUD  2048
#define NCAT    4096
#define NPOINT  1024
#define FPS_T   256
#define PPT     (NCAT / FPS_T)   // 16 points per thread

// ---------------------------------------------------------------------------
// Farthest point sampling: one workgroup per batch. Each thread owns 16 points
// (coords + running min-distance) in registers. Per iteration: broadcast
// centroid (uniform global load), update dists, wave32 argmax via shfl_xor,
// then cross-wave argmax through LDS. Tie-breaking prefers the smallest global
// index to match jnp.argmax. Emits coarse_pc[it] = concat_pc[far] directly.
// ---------------------------------------------------------------------------
__global__ __launch_bounds__(FPS_T)
void fps_kernel(const float* __restrict__ recon, const float* __restrict__ part,
                float* __restrict__ coarse /* [B,NPOINT,3] */) {
  const int b = blockIdx.x;
  const int t = threadIdx.x;
  const float* rb = recon + (size_t)b * NCLOUD * 3;
  const float* pb = part  + (size_t)b * NCLOUD * 3;
  float* cb = coarse + (size_t)b * NPOINT * 3;

  float px[PPT], py[PPT], pz[PPT], dist[PPT];
#pragma unroll
  for (int k = 0; k < PPT; ++k) {
    const int i = k * FPS_T + t;
    const float* s = (i < NCLOUD) ? (rb + 3 * i) : (pb + 3 * (i - NCLOUD));
    px[k] = s[0]; py[k] = s[1]; pz[k] = s[2];
    dist[k] = 1e30f;
  }

  __shared__ float s_val[FPS_T / 32];
  __shared__ int   s_idx[FPS_T / 32];
  __shared__ int   s_far;
  if (t == 0) s_far = 0;
  __syncthreads();

  for (int it = 0; it < NPOINT; ++it) {
    const int far = s_far;
    const float* cp = (far < NCLOUD) ? (rb + 3 * far) : (pb + 3 * (far - NCLOUD));
    const float cx = cp[0], cy = cp[1], cz = cp[2];

    float best = -1.0f; int bidx = 0;
#pragma unroll
    for (int k = 0; k < PPT; ++k) {
      const float dx = px[k] - cx, dy = py[k] - cy, dz = pz[k] - cz;
      const float d  = dx * dx + dy * dy + dz * dz;
      const float nd = fminf(dist[k], d);
      dist[k] = nd;
      if (nd > best) { best = nd; bidx = k * FPS_T + t; }
    }
    // wave32 argmax (prefer smaller index on ties)
#pragma unroll
    for (int m = 16; m > 0; m >>= 1) {
      const float ov = __shfl_xor(best, m, 32);
      const int   oi = __shfl_xor(bidx, m, 32);
      if (ov > best || (ov == best && oi < bidx)) { best = ov; bidx = oi; }
    }
    if ((t & 31) == 0) { s_val[t >> 5] = best; s_idx[t >> 5] = bidx; }
    __syncthreads();
    if (t == 0) {
      float bv = s_val[0]; int bi = s_idx[0];
      for (int w = 1; w < FPS_T / 32; ++w) {
        if (s_val[w] > bv || (s_val[w] == bv && s_idx[w] < bi)) {
          bv = s_val[w]; bi = s_idx[w];
        }
      }
      s_far = bi;
      cb[3 * it + 0] = cx; cb[3 * it + 1] = cy; cb[3 * it + 2] = cz;
    }
    __syncthreads();
  }
}

// ---------------------------------------------------------------------------
// One-sided chamfer min via V_WMMA_F32_16X16X4_F32.
// One wave owns a 16-row a-tile; A = 16 x (x,y,z,0). For each 16-col b-tile,
// B = (x,y,z,0)^T x 16, one WMMA gives the 16x16 inner-product tile;
// sq = aa + bb - 2*ab, running min in 8 accumulators (C layout: VGPR v,
// lane l -> M = v + 8*(l>>4), N = l&15). Final min across the 16-lane group
// via xor shuffles; lanes 0 / 16 write the 16 per-point mins.
// ---------------------------------------------------------------------------
__global__ __launch_bounds__(256)
void chamfer_min_kernel(const float* __restrict__ apts, int aStride,
                        const float* __restrict__ bpts, int bStride,
                        int nATiles, int nBTiles,
                        float* __restrict__ out, int outStride) {
  const int wid  = blockIdx.x * (blockDim.x >> 5) + (threadIdx.x >> 5);
  const int lane = threadIdx.x & 31;
  const int nWaves = BATCH * nATiles;
  if (wid >= nWaves) return;                 // uniform per wave
  const int b     = wid / nATiles;
  const int atile = wid % nATiles;
  const int hi    = lane >> 4;               // half-wave: K={0,1} vs K={2,3}
  const int ln    = lane & 15;

  const float* abase = apts + (size_t)b * aStride;
  const float* bbase = bpts + (size_t)b * bStride;

  const float* ap = abase + 3 * (atile * 16 + ln);
  const float ax = ap[0], ay = ap[1], az = ap[2];
  v2f A;
  A.x = hi ? az : ax;
  A.y = hi ? 0.0f : ay;
  const float aa_lane = ax * ax + ay * ay + az * az;
  float aaM[8];
#pragma unroll
  for (int v = 0; v < 8; ++v) aaM[v] = __shfl(aa_lane, v + 8 * hi, 32);

  float mn[8];
#pragma unroll
  for (int v = 0; v < 8; ++v) mn[v] = 1e30f;

  for (int bt = 0; bt < nBTiles; ++bt) {
    const float* bp = bbase + 3 * (bt * 16 + ln);
    const float bx = bp[0], by = bp[1], bz = bp[2];
    v2f Bm;
    Bm.x = hi ? bz : bx;
    Bm.y = hi ? 0.0f : by;
    const float bbv = bx * bx + by * by + bz * bz;
    v8f C = {};
    // D = A(16x4) x B(4x16): pairwise dot products over (x,y,z,0)
    v8f D = __builtin_amdgcn_wmma_f32_16x16x4_f32(
        false, A, false, Bm, (short)0, C, false, false);
#pragma unroll
    for (int v = 0; v < 8; ++v) {
      const float sq = aaM[v] + bbv - 2.0f * D[v];
      mn[v] = fminf(mn[v], sq);
    }
  }
  // min across the 16 lanes of each half-wave group (xor<16 stays in-group)
#pragma unroll
  for (int v = 0; v < 8; ++v) {
#pragma unroll
    for (int m = 1; m < 16; m <<= 1) mn[v] = fminf(mn[v], __shfl_xor(mn[v], m, 32));
  }
  if (ln == 0) {
    float* o = out + (size_t)b * outStride + atile * 16 + 8 * hi;
#pragma unroll
    for (int v = 0; v < 8; ++v) o[v] = mn[v];
  }
}

// ---------------------------------------------------------------------------
// Deterministic tree sum of the B*512 theta partial mins -> theta scalar.
// ---------------------------------------------------------------------------
__global__ __launch_bounds__(256)
void theta_reduce_kernel(const float* __restrict__ parts, float* __restrict__ theta) {
  __shared__ float s[256];
  const int t = threadIdx.x;
  float acc = 0.0f;
  for (int i = t; i < BATCH * 512; i += 256) acc += parts[i];
  s[t] = acc;
  __syncthreads();
  for (int off = 128; off > 0; off >>= 1) {
    if (t < off) s[t] += s[t + off];
    __syncthreads();
  }
  if (t == 0) theta[0] = s[0] / (float)(BATCH * 512);
}

// ---------------------------------------------------------------------------
// fine = tile(coarse,2) + (masked-clamped) offset; passthrough reconstructed.
// ---------------------------------------------------------------------------
__global__ __launch_bounds__(256)
void finalize_kernel(const float* __restrict__ recon, const float* __restrict__ offset,
                     const float* __restrict__ coarse, const float* __restrict__ dcp,
                     const float* __restrict__ theta_p,
                     float* __restrict__ fine, float* __restrict__ recon_out) {
  const int gid = blockIdx.x * blockDim.x + threadIdx.x;
  if (gid >= BATCH * 2 * NPOINT) return;
  const int b = gid / (2 * NPOINT);
  const int j = gid % (2 * NPOINT);
  const float theta = theta_p[0];
  const int  cj   = j & (NPOINT - 1);
  const bool mask = (j < NPOINT) && (dcp[(size_t)b * NPOINT + j] <= theta);
  const float* c   = coarse + ((size_t)b * NPOINT + cj) * 3;
  const float* off = offset + (size_t)gid * 3;
  float* fo = fine + (size_t)gid * 3;
#pragma unroll
  for (int d = 0; d < 3; ++d) {
    float o = off[d];
    if (mask) o = fminf(fmaxf(o, -0.02f), 0.02f);
    fo[d] = c[d] + o;
  }
#pragma unroll
  for (int d = 0; d < 3; ++d)
    recon_out[(size_t)gid * 3 + d] = recon[(size_t)gid * 3 + d];
}

extern "C" void kernel_launch(void* const* d_in, const int* in_sizes, int n_in,
                              void* d_out, int out_size, void* d_ws, size_t ws_size,
                              hipStream_t stream) {
  const float* partial = (const float*)d_in[0];  // [B,2048,3]
  const float* recon   = (const float*)d_in[1];  // [B,2048,3]
  const float* offset  = (const float*)d_in[2];  // [B,2048,3]

  float* out       = (float*)d_out;
  float* fine      = out;                               // [B,2048,3]
  float* recon_out = out + BATCH * NCLOUD * 3;          // [B,2048,3]
  float* coarse    = recon_out + BATCH * NCLOUD * 3;    // [B,1024,3]

  float* ws          = (float*)d_ws;
  float* dcp         = ws;                              // [B,1024]
  float* theta_parts = dcp + BATCH * NPOINT;            // [B,512]
  float* theta       = theta_parts + BATCH * 512;       // [1]

  // 1) FPS -> coarse_pc (also an output)
  fps_kernel<<<BATCH, FPS_T, 0, stream>>>(recon, partial, coarse);

  // 2) theta partials: chamfer(coarse[:, :512], coarse[:, 512:])
  {
    const int nATiles = 512 / 16, nBTiles = 512 / 16;
    const int nWaves = BATCH * nATiles;
    chamfer_min_kernel<<<(nWaves + 7) / 8, 256, 0, stream>>>(
        coarse, NPOINT * 3, coarse + 512 * 3, NPOINT * 3,
        nATiles, nBTiles, theta_parts, 512);
  }

  // 3) theta scalar (deterministic reduction)
  theta_reduce_kernel<<<1, 256, 0, stream>>>(theta_parts, theta);

  // 4) d_cp = chamfer(coarse, partial)
  {
    const int nATiles = NPOINT / 16, nBTiles = NCLOUD / 16;
    const int nWaves = BATCH * nATiles;
    chamfer_min_kernel<<<(nWaves + 7) / 8, 256, 0, stream>>>(
        coarse, NPOINT * 3, partial, NCLOUD * 3,
        nATiles, nBTiles, dcp, NPOINT);
  }

  // 5) final assembly + reconstructed passthrough
  {
    const int total = BATCH * 2 * NPOINT;
    finalize_kernel<<<(total + 255) / 256, 256, 0, stream>>>(
        recon, offset, coarse, dcp, theta, fine, recon_out);
  }
}